// GuidedFilter_42408507080840
// MI455X (gfx1250) — compile-verified
//
#include <hip/hip_runtime.h>

// Guided filter for MI455X (gfx1250, wave32).
// Box filter = causal trailing window of size r (=20) along H then W.
// Memory-bound (~1.2 GB moved -> ~60us @ 23.3 TB/s). Horizontal window sums
// run on the matrix pipe as banded-matrix WMMA (V_WMMA_F32_16X16X4_F32)
// with exact 0/1 f32 weights.
//
// Round-3 changes: kill residual scratch spills in hbox kernels via
// __launch_bounds__(256,1) (occupancy is wave-count limited anyway: 1536
// waves total), hoist J-invariant address math, and cut 16 precise-division
// sequences per tile down to 1 (invN = inv_ni[v] * inv_nj).

typedef __attribute__((ext_vector_type(2))) float v2f;
typedef __attribute__((ext_vector_type(8))) float v8f;

#define HDIM 1024
#define WDIM 1024
#define HW   (HDIM * WDIM)
#define NIMG 24            // B*C = 8*3
#define NELEM ((size_t)NIMG * HW)
#define SEGS 4
#define SEG_ROWS (HDIM / SEGS)

static __device__ __forceinline__ v8f wmma_f32(v2f a, v2f b, v8f c) {
  // D = A(16x4) * B(4x16) + C(16x16), fp32 matrix pipe
  return __builtin_amdgcn_wmma_f32_16x16x4_f32(false, a, false, b, (short)0, c,
                                               false, false);
}

static __device__ __forceinline__ int imin(int a, int b) { return a < b ? a : b; }

// ---------------- Vertical trailing box, 4 quantities (I, p, I*p, I*I) -----
__global__ void vbox4_kernel(const float* __restrict__ I, const float* __restrict__ P,
                             float* __restrict__ t0, float* __restrict__ t1,
                             float* __restrict__ t2, float* __restrict__ t3,
                             const int* __restrict__ rp) {
  const int r = *rp;
  int tid = blockIdx.x * blockDim.x + threadIdx.x;
  const int NCOL = NIMG * WDIM;
  int seg = tid / NCOL;
  int c   = tid - seg * NCOL;
  int img = c >> 10;
  int j   = c & (WDIM - 1);
  size_t base = (size_t)img * HW + j;
  int i0 = seg * SEG_ROWS;
  int iw = i0 - (r - 1); if (iw < 0) iw = 0;
  float sI = 0.f, sP = 0.f, sIp = 0.f, sII = 0.f;
  for (int i = iw; i < i0 + SEG_ROWS; ++i) {
    size_t idx = base + (size_t)i * WDIM;
    float a = I[idx], b = P[idx];
    sI += a; sP += b; sIp += a * b; sII += a * a;
    int k = i - r;
    if (k >= iw) {                       // remove element leaving the window
      size_t kx = base + (size_t)k * WDIM;
      float ao = I[kx], bo = P[kx];
      sI -= ao; sP -= bo; sIp -= ao * bo; sII -= ao * ao;
    }
    if (i >= i0) { t0[idx] = sI; t1[idx] = sP; t2[idx] = sIp; t3[idx] = sII; }
  }
}

// ---------------- Vertical trailing box, 2 quantities (A, b) ---------------
__global__ void vbox2_kernel(const float* __restrict__ A, const float* __restrict__ Bv,
                             float* __restrict__ tA, float* __restrict__ tB,
                             const int* __restrict__ rp) {
  const int r = *rp;
  int tid = blockIdx.x * blockDim.x + threadIdx.x;
  const int NCOL = NIMG * WDIM;
  int seg = tid / NCOL;
  int c   = tid - seg * NCOL;
  int img = c >> 10;
  int j   = c & (WDIM - 1);
  size_t base = (size_t)img * HW + j;
  int i0 = seg * SEG_ROWS;
  int iw = i0 - (r - 1); if (iw < 0) iw = 0;
  float sA = 0.f, sB = 0.f;
  for (int i = iw; i < i0 + SEG_ROWS; ++i) {
    size_t idx = base + (size_t)i * WDIM;
    sA += A[idx]; sB += Bv[idx];
    int k = i - r;
    if (k >= iw) {
      size_t kx = base + (size_t)k * WDIM;
      sA -= A[kx]; sB -= Bv[kx];
    }
    if (i >= i0) { tA[idx] = sA; tB[idx] = sB; }
  }
}

// Banded weight fragments: block m covers input cols J-32+16m .. +15.
// diff = (out col) - (in col) = n + 32 - 16m - k ; weight = (0 <= diff < r).
// B 4x16 layout (mirrors documented A 16x4): lanes 0-15 hold K={0,1},
// lanes 16-31 hold K={2,3}; .x = even K, .y = odd K; N = lane&15.
static __device__ __forceinline__ void make_bands(int n, int kq, int r, v2f Bf[3][4]) {
#pragma unroll
  for (int m = 0; m < 3; ++m) {
#pragma unroll
    for (int s = 0; s < 4; ++s) {
      int k0 = 4 * s + 2 * kq;
      int d0 = n + 32 - 16 * m - k0;
      int d1 = d0 - 1;
      Bf[m][s].x = (d0 >= 0 && d0 < r) ? 1.0f : 0.0f;
      Bf[m][s].y = (d1 >= 0 && d1 < r) ? 1.0f : 0.0f;
    }
  }
}

// ---------------- Horizontal WMMA box, 4 quantities -> A, b (in place) -----
// One wave owns a 16-row strip and slides J by 16.
//  * I,p sums (t0,t1) are overwritten with A,b => keep a 3-block register
//    ring so old columns are never re-read (race-free in-place).
//  * Ip,II sums (t2,t3) are read-only => re-read all 3 blocks each step
//    (old blocks are L0-hot).
__global__ void __launch_bounds__(256, 1)
hbox4_kernel(float* __restrict__ t0, float* __restrict__ t1,
             const float* __restrict__ t2, const float* __restrict__ t3,
             const int* __restrict__ rp) {
  const int r = *rp;
  int lane = threadIdx.x;
  int n  = lane & 15;
  int kq = lane >> 4;
  int strip = blockIdx.x * blockDim.y + threadIdx.y;  // 24*64 strips
  int img = strip >> 6;
  int R   = (strip & 63) << 4;

  // A-frag load base: row = R+n, 2*kq folded into the pointer.
  size_t lbase = (size_t)img * HW + (size_t)(R + n) * WDIM + (size_t)(2 * kq);
  const float* pI  = t0 + lbase;
  const float* pP  = t1 + lbase;
  const float* pIp = t2 + lbase;
  const float* pII = t3 + lbase;
  // Store base: row block folded, col = J + n added in-loop.
  size_t sbase = (size_t)img * HW + (size_t)R * WDIM + n;

  v2f Bf[3][4];
  make_bands(n, kq, r, Bf);

  float inv_ni[8];                      // 1 / min(row+1, r), J-invariant
#pragma unroll
  for (int v = 0; v < 8; ++v)
    inv_ni[v] = 1.0f / (float)imin(R + v + 8 * kq + 1, r);

  v2f fI[3][4], fP[3][4];               // register ring (in-place bufs)
#pragma unroll
  for (int m = 0; m < 3; ++m)
#pragma unroll
    for (int s = 0; s < 4; ++s) { fI[m][s] = (v2f)0.0f; fP[m][s] = (v2f)0.0f; }

  for (int J = 0; J < WDIM; J += 16) {
#pragma unroll
    for (int s = 0; s < 4; ++s) {       // ring shift
      fI[0][s] = fI[1][s];  fI[1][s] = fI[2][s];
      fP[0][s] = fP[1][s];  fP[1][s] = fP[2][s];
    }
#pragma unroll
    for (int s = 0; s < 4; ++s) {       // load newest block (cols J..J+15)
      fI[2][s] = *(const v2f*)(pI + J + 4 * s);
      fP[2][s] = *(const v2f*)(pP + J + 4 * s);
    }

    v8f aI = {}, aP = {}, aIp = {}, aII = {};
#pragma unroll
    for (int m = 0; m < 3; ++m) {
#pragma unroll
      for (int s = 0; s < 4; ++s) {
        aI = wmma_f32(fI[m][s], Bf[m][s], aI);
        aP = wmma_f32(fP[m][s], Bf[m][s], aP);
      }
      int cb = J - 32 + 16 * m;         // uniform guard -> scalar branch
      v2f gIp[4], gII[4];
      if (cb >= 0) {
#pragma unroll
        for (int s = 0; s < 4; ++s) {
          gIp[s] = *(const v2f*)(pIp + cb + 4 * s);
          gII[s] = *(const v2f*)(pII + cb + 4 * s);
        }
      } else {
#pragma unroll
        for (int s = 0; s < 4; ++s) { gIp[s] = (v2f)0.0f; gII[s] = (v2f)0.0f; }
      }
#pragma unroll
      for (int s = 0; s < 4; ++s) {
        aIp = wmma_f32(gIp[s], Bf[m][s], aIp);
        aII = wmma_f32(gII[s], Bf[m][s], aII);
      }
    }

    float inv_nj = 1.0f / (float)imin(J + n + 1, r);   // one div per tile
#pragma unroll
    for (int v = 0; v < 8; ++v) {       // C/D layout: M = v + 8*kq, N = n
      float invN = inv_ni[v] * inv_nj;
      float mI = aI[v] * invN, mp = aP[v] * invN;
      float cov = aIp[v] * invN - mI * mp;
      float var = aII[v] * invN - mI * mI;
      float Aq = cov / (var + 1e-8f);
      float bq = mp - Aq * mI;
      size_t pos = sbase + (size_t)((v + 8 * kq) << 10) + J;
      t0[pos] = Aq;
      t1[pos] = bq;
    }
  }
}

// ---------------- Horizontal WMMA box on (A,b) sums + final q --------------
__global__ void __launch_bounds__(256, 1)
hbox2_final_kernel(const float* __restrict__ tA, const float* __restrict__ tB,
                   const float* __restrict__ I, float* __restrict__ q,
                   const int* __restrict__ rp) {
  const int r = *rp;
  int lane = threadIdx.x;
  int n  = lane & 15;
  int kq = lane >> 4;
  int strip = blockIdx.x * blockDim.y + threadIdx.y;
  int img = strip >> 6;
  int R   = (strip & 63) << 4;

  size_t lbase = (size_t)img * HW + (size_t)(R + n) * WDIM + (size_t)(2 * kq);
  const float* pA = tA + lbase;
  const float* pB = tB + lbase;
  size_t sbase = (size_t)img * HW + (size_t)R * WDIM + n;

  v2f Bf[3][4];
  make_bands(n, kq, r, Bf);

  float inv_ni[8];
#pragma unroll
  for (int v = 0; v < 8; ++v)
    inv_ni[v] = 1.0f / (float)imin(R + v + 8 * kq + 1, r);

  v2f fA[3][4], fB[3][4];
#pragma unroll
  for (int m = 0; m < 3; ++m)
#pragma unroll
    for (int s = 0; s < 4; ++s) { fA[m][s] = (v2f)0.0f; fB[m][s] = (v2f)0.0f; }

  for (int J = 0; J < WDIM; J += 16) {
#pragma unroll
    for (int s = 0; s < 4; ++s) {
      fA[0][s] = fA[1][s]; fA[1][s] = fA[2][s];
      fB[0][s] = fB[1][s]; fB[1][s] = fB[2][s];
    }
#pragma unroll
    for (int s = 0; s < 4; ++s) {
      fA[2][s] = *(const v2f*)(pA + J + 4 * s);
      fB[2][s] = *(const v2f*)(pB + J + 4 * s);
    }
    v8f aA = {}, aB = {};
#pragma unroll
    for (int m = 0; m < 3; ++m)
#pragma unroll
      for (int s = 0; s < 4; ++s) {
        aA = wmma_f32(fA[m][s], Bf[m][s], aA);
        aB = wmma_f32(fB[m][s], Bf[m][s], aB);
      }
    float inv_nj = 1.0f / (float)imin(J + n + 1, r);
#pragma unroll
    for (int v = 0; v < 8; ++v) {
      float invN = inv_ni[v] * inv_nj;
      size_t pos = sbase + (size_t)((v + 8 * kq) << 10) + J;
      q[pos] = (aA[v] * invN) * I[pos] + (aB[v] * invN);
    }
  }
}

extern "C" void kernel_launch(void* const* d_in, const int* in_sizes, int n_in,
                              void* d_out, int out_size, void* d_ws, size_t ws_size,
                              hipStream_t stream) {
  const float* I = (const float*)d_in[0];
  const float* P = (const float*)d_in[1];
  const int*   rp = (const int*)d_in[2];
  float* out = (float*)d_out;

  float* T0 = (float*)d_ws;          // vbox(I)   -> later A
  float* T1 = T0 + NELEM;            // vbox(p)   -> later b
  float* T2 = T1 + NELEM;            // vbox(I*p) -> later vbox(A)
  float* T3 = T2 + NELEM;            // vbox(I*I) -> later vbox(b)

  (void)in_sizes; (void)n_in; (void)out_size; (void)ws_size;

  // Stage 1: vertical trailing box of I, p, I*p, I*I
  vbox4_kernel<<<(NIMG * WDIM * SEGS) / 256, 256, 0, stream>>>(I, P, T0, T1, T2, T3, rp);
  // Stage 2: horizontal box via WMMA; compute A,b (in place over T0,T1)
  hbox4_kernel<<<(NIMG * 64) / 8, dim3(32, 8), 0, stream>>>(T0, T1, T2, T3, rp);
  // Stage 3: vertical trailing box of A, b
  vbox2_kernel<<<(NIMG * WDIM * SEGS) / 256, 256, 0, stream>>>(T0, T1, T2, T3, rp);
  // Stage 4: horizontal box via WMMA + q = mean_A * I + mean_b
  hbox2_final_kernel<<<(NIMG * 64) / 8, dim3(32, 8), 0, stream>>>(T2, T3, I, out, rp);
}